// GramSchmidt04_19980187861912
// MI455X (gfx1250) — compile-verified
//
#include <hip/hip_runtime.h>
#include <hip/hip_bf16.h>

typedef float v2f __attribute__((ext_vector_type(2)));
typedef float v8f __attribute__((ext_vector_type(8)));

#define KROWS 64
#define LCOLS 512
#define XS 516      // padded LDS row stride (floats): bank offset 4/row -> conflict-free tiles
#define ZSTR 52     // zbuf row stride (floats)
#define CSTR 18     // cred row stride
#define CREDW (16*CSTR)
#define NW 8        // waves per block (256 threads, wave32)

__global__ __launch_bounds__(256, 1)
void GramSchmidt_gfx1250_kernel(const float* __restrict__ xin,
                                float* __restrict__ xout) {
    __shared__ float X[KROWS * XS];        // 132096 B
    __shared__ float zbuf[16 * ZSTR];      // 3328 B
    __shared__ float cred[NW * CREDW];     // 9216 B
    __shared__ float zs[16];
    __shared__ float norms[KROWS];
    __shared__ float invn[KROWS];
    __shared__ float wred[NW];

    const int t    = threadIdx.x;
    const int wave = t >> 5;
    const int lane = t & 31;
    const int lr   = lane & 15;
    const int half = lane >> 4;
    const long pbase = (long)blockIdx.x * (KROWS * LCOLS);

    // ---- load the whole 64x512 fp32 tile into LDS (single pass) ----
    for (int i = 0; i < 32; ++i) {
        int f   = t * 4 + i * 1024;
        int row = f >> 9, col = f & 511;
        const float4 v = *reinterpret_cast<const float4*>(xin + pbase + f);
        *reinterpret_cast<float4*>(&X[row * XS + col]) = v;
    }
    __syncthreads();

    for (int b = 0; b < 4; ++b) {
        const int r0 = b * 16;

        if (b > 0) {
            // ---- Z1 = X_b (16xL) * X_prev^T, one 16x16 tile per prev block,
            //      split-K over the 8 waves, WMMA f32 16x16x4 ----
            for (int p = 0; p < b; ++p) {
                v8f acc = {};
                for (int s = 0; s < 16; ++s) {
                    const int l0 = 4 * (wave + 8 * s);   // union over waves = all of [0,512) step 4
                    v2f a, bb;
                    a.x  = X[(r0 + lr) * XS + l0 + 2 * half];
                    a.y  = X[(r0 + lr) * XS + l0 + 2 * half + 1];
                    bb.x = X[(p * 16 + lr) * XS + l0 + 2 * half];
                    bb.y = X[(p * 16 + lr) * XS + l0 + 2 * half + 1];
                    acc = __builtin_amdgcn_wmma_f32_16x16x4_f32(
                        false, a, false, bb, (short)0, acc, false, false);
                }
                #pragma unroll
                for (int vr = 0; vr < 8; ++vr) {
                    const int m = vr + 8 * half;          // C/D layout: M = vr + 8*half, N = lr
                    cred[wave * CREDW + m * CSTR + lr] = acc[vr];
                }
                __syncthreads();
                {   // cross-wave reduce + scale by 1/||x_k||^2
                    const int m = t >> 4, n = t & 15;
                    float s = 0.f;
                    #pragma unroll
                    for (int w = 0; w < NW; ++w) s += cred[w * CREDW + m * CSTR + n];
                    zbuf[m * ZSTR + p * 16 + n] = s * invn[p * 16 + n];
                }
                __syncthreads();
            }

            // ---- Proj = Z (16 x r0) * X_prev (r0 x 512); split-N over waves ----
            for (int q = 0; q < 4; ++q) {
                const int n0 = 16 * (wave + 8 * q);
                v8f acc = {};
                for (int k0 = 0; k0 < r0; k0 += 4) {
                    v2f a, bb;
                    a.x  = zbuf[lr * ZSTR + k0 + 2 * half];
                    a.y  = zbuf[lr * ZSTR + k0 + 2 * half + 1];
                    bb.x = X[(k0 + 2 * half) * XS + n0 + lr];
                    bb.y = X[(k0 + 2 * half + 1) * XS + n0 + lr];
                    acc = __builtin_amdgcn_wmma_f32_16x16x4_f32(
                        false, a, false, bb, (short)0, acc, false, false);
                }
                #pragma unroll
                for (int vr = 0; vr < 8; ++vr) {
                    const int m = vr + 8 * half;
                    X[(r0 + m) * XS + n0 + lr] -= acc[vr];   // unique (row,col) per lane
                }
            }
            __syncthreads();
        }

        // ---- intra-block sequential Gram-Schmidt (16 steps) ----
        for (int ii = 0; ii < 16; ++ii) {
            const int i = r0 + ii;
            if (ii > 0) {
                for (int kk = wave; kk < ii; kk += NW) {
                    const int krow = r0 + kk;
                    float d = 0.f;
                    #pragma unroll
                    for (int u = 0; u < 16; ++u) {
                        const int c = lane + 32 * u;
                        d += X[i * XS + c] * X[krow * XS + c];
                    }
                    #pragma unroll
                    for (int off = 16; off > 0; off >>= 1) d += __shfl_xor(d, off, 32);
                    if (lane == 0) zs[kk] = d * invn[krow];
                }
                __syncthreads();
                float a0 = 0.f, a1 = 0.f;
                for (int kk = 0; kk < ii; ++kk) {
                    const float zv = zs[kk];
                    a0 += zv * X[(r0 + kk) * XS + t];
                    a1 += zv * X[(r0 + kk) * XS + t + 256];
                }
                X[i * XS + t]       -= a0;
                X[i * XS + t + 256] -= a1;
                __syncthreads();
            }
            // ---- norm of finalized row i (= z2 for later steps) ----
            {
                const float x0 = X[i * XS + t], x1 = X[i * XS + t + 256];
                float sq = x0 * x0 + x1 * x1;
                #pragma unroll
                for (int off = 16; off > 0; off >>= 1) sq += __shfl_xor(sq, off, 32);
                if (lane == 0) wred[wave] = sq;
                __syncthreads();
                if (t == 0) {
                    float s = 0.f;
                    #pragma unroll
                    for (int w = 0; w < NW; ++w) s += wred[w];
                    norms[i] = s;
                    invn[i]  = 1.0f / s;
                }
                __syncthreads();
            }
        }
    }

    // ---- final normalization + single-pass writeback ----
    for (int i = 0; i < 32; ++i) {
        const int f   = t * 4 + i * 1024;
        const int row = f >> 9, col = f & 511;
        const float sc = rsqrtf(norms[row]);
        float4 v = *reinterpret_cast<const float4*>(&X[row * XS + col]);
        v.x *= sc; v.y *= sc; v.z *= sc; v.w *= sc;
        *reinterpret_cast<float4*>(xout + pbase + f) = v;
    }
}

extern "C" void kernel_launch(void* const* d_in, const int* in_sizes, int n_in,
                              void* d_out, int out_size, void* d_ws, size_t ws_size,
                              hipStream_t stream) {
    (void)in_sizes; (void)n_in; (void)d_ws; (void)ws_size; (void)out_size;
    const float* x = (const float*)d_in[0];
    float* out = (float*)d_out;
    GramSchmidt_gfx1250_kernel<<<dim3(32 * 32), dim3(256), 0, stream>>>(x, out);
}